// QuantumClassifier_34316788695640
// MI455X (gfx1250) — compile-verified
//
#include <hip/hip_runtime.h>
#include <math.h>

// CDNA5 / gfx1250: wave32, WMMA f32 16x16x4.
typedef __attribute__((ext_vector_type(2))) float v2f;
typedef __attribute__((ext_vector_type(8))) float v8f;

#define BLOCK          256
#define ROWS_PER_WAVE  32
#define ROWS_PER_BLOCK 256   // 8 waves * 32 rows
#define FDIM           96
#define KCHUNKS        24    // 96 / 4

// Fast tanh via hardware v_exp_f32: tanh(x) = sign(x) * (1 - t) / (1 + t),
// t = exp(-2|x|). Accurate to ~1e-6 rel for all x, no range reduction needed.
__device__ __forceinline__ float fast_tanhf(float v) {
    const float ax = fabsf(v);
    const float t  = __expf(-2.0f * ax);
    const float r  = (1.0f - t) / (1.0f + t);
    return copysignf(r, v);
}

__global__ __launch_bounds__(BLOCK)
void qc_fused_kernel(const float* __restrict__ x,
                     const float* __restrict__ Wpre,   // (4, 96)
                     const float* __restrict__ bpre,   // (4,)
                     const float* __restrict__ qp,     // (4, 3) phi,theta,omega
                     const float* __restrict__ Wpost,  // (2, 4)
                     const float* __restrict__ bpost,  // (2,)
                     float* __restrict__ out,          // (B, 2)
                     int B)
{
    // B-matrix (W_pre^T) padded to 16 N-rows (rows 4..15 zero) so WMMA needs
    // no predication and EXEC stays all-ones through the MMA loop.
    __shared__ float w_lds[16 * FDIM];            // 6 KB
    __shared__ float ang[ROWS_PER_BLOCK * 4];     // 4 KB: per-row pre-activations

    const int tid  = threadIdx.x;
    const int lane = tid & 31;
    const int wave = tid >> 5;

    for (int i = tid; i < 16 * FDIM; i += BLOCK)
        w_lds[i] = (i < 4 * FDIM) ? Wpre[i] : 0.0f;
    __syncthreads();

    const int h = lane >> 4;          // half-wave: selects K pair {2h, 2h+1}
    const int n = lane & 15;          // N index (qubit for n<4)
    const int rowbase = blockIdx.x * ROWS_PER_BLOCK + wave * ROWS_PER_WAVE;

    // Clamp A-row reads for a possible ragged last block (stores are guarded).
    int r0 = rowbase + n;            if (r0 >= B) r0 = B - 1;
    int r1 = rowbase + 16 + n;       if (r1 >= B) r1 = B - 1;

    const float* pa0 = x + (size_t)r0 * FDIM + 2 * h;   // A tile 0 (M=0..15)
    const float* pa1 = x + (size_t)r1 * FDIM + 2 * h;   // A tile 1 (M=16..31)
    const float* pb  = w_lds + n * FDIM + 2 * h;        // B fragment source

    v8f acc0 = {0.f,0.f,0.f,0.f,0.f,0.f,0.f,0.f};
    v8f acc1 = {0.f,0.f,0.f,0.f,0.f,0.f,0.f,0.f};

#pragma unroll
    for (int t = 0; t < KCHUNKS; ++t) {
        v2f a0 = *(const v2f*)(pa0 + 4 * t);   // global_load_b64
        v2f a1 = *(const v2f*)(pa1 + 4 * t);   // global_load_b64
        v2f b  = *(const v2f*)(pb  + 4 * t);   // ds_load_b64 (pairs merge to 2addr)
        acc0 = __builtin_amdgcn_wmma_f32_16x16x4_f32(false, a0, false, b,
                                                     (short)0, acc0, false, false);
        acc1 = __builtin_amdgcn_wmma_f32_16x16x4_f32(false, a1, false, b,
                                                     (short)0, acc1, false, false);
    }

    // Redistribute: D layout is (M = r + 8h, N = lane&15). Lanes with n<4 own
    // the 4 real output columns; scatter through LDS so each lane gets one row.
    if (n < 4) {
        const float bp = bpre[n];
#pragma unroll
        for (int r = 0; r < 8; ++r) {
            const int m = r + 8 * h;
            ang[(wave * ROWS_PER_WAVE + m) * 4 + n]      = acc0[r] + bp;
            ang[(wave * ROWS_PER_WAVE + 16 + m) * 4 + n] = acc1[r] + bp;
        }
    }
    __syncthreads();

    // Closed-form epilogue. Quantum circuit reduces to:
    //   a_w = pi * tanh(acc_w)
    //   d_w = cos(theta_w)cos(a_w) - sin(theta_w)cos(phi_w)sin(a_w)
    //   z   = cumprod(d)   (CNOT ladder -> parity products; omega drops out)
    const int row = rowbase + lane;
    if (row < B) {
        float av[4];
#pragma unroll
        for (int w = 0; w < 4; ++w)
            av[w] = ang[(wave * ROWS_PER_WAVE + lane) * 4 + w];   // ds_load_b128

        float z = 1.0f, zo[4];
#pragma unroll
        for (int w = 0; w < 4; ++w) {
            const float phi = qp[3 * w + 0];
            const float th  = qp[3 * w + 1];
            const float a   = 3.14159265358979323846f * fast_tanhf(av[w]);
            // hardware v_cos_f32 / v_sin_f32; args are in (-pi, pi) or ~N(0,0.1)
            const float d   = __cosf(th) * __cosf(a)
                            - __sinf(th) * __cosf(phi) * __sinf(a);
            z *= d;
            zo[w] = z;
        }

        float o0 = bpost[0], o1 = bpost[1];
#pragma unroll
        for (int w = 0; w < 4; ++w) {
            o0 += Wpost[w]     * zo[w];
            o1 += Wpost[4 + w] * zo[w];
        }
        v2f o = {o0, o1};
        *(v2f*)(out + (size_t)row * 2) = o;   // coalesced global_store_b64
    }
}

extern "C" void kernel_launch(void* const* d_in, const int* in_sizes, int n_in,
                              void* d_out, int out_size, void* d_ws, size_t ws_size,
                              hipStream_t stream) {
    const float* x     = (const float*)d_in[0];
    const float* Wpre  = (const float*)d_in[1];
    const float* bpre  = (const float*)d_in[2];
    const float* qp    = (const float*)d_in[3];
    const float* Wpost = (const float*)d_in[4];
    const float* bpost = (const float*)d_in[5];
    float* out = (float*)d_out;

    const int B = in_sizes[0] / FDIM;               // 262144
    const int grid = (B + ROWS_PER_BLOCK - 1) / ROWS_PER_BLOCK;   // 1024

    qc_fused_kernel<<<grid, BLOCK, 0, stream>>>(x, Wpre, bpre, qp, Wpost, bpost,
                                                out, B);
}